// GACN_32341103739239
// MI455X (gfx1250) — compile-verified
//
#include <hip/hip_runtime.h>
#include <cstdint>
#include <cstddef>

// LightGCN forward on MI455X (gfx1250).
//
// Roofline: 2.46 GFLOP total (WMMA irrelevant); bound by
//   - edge streaming: 3 x 76.8 MB from HBM (~10us at 23.3 TB/s)
//   - L2 gather/atomic throughput: x (51.2MB) and y (51.2MB) fit in 192MB L2
// Strategy:
//   - edges streamed with gfx1250 async-to-LDS b128 (ASYNCcnt), double
//     buffered per wave (wave-private LDS slices => no barriers)
//   - 16 lanes/edge: one coalesced 256B row gather (b128/lane) + 4 fp32
//     hardware atomics per lane into y
//
// k_atomic_probe is defined FIRST purely so the compile-loop disasm snippet
// shows the exact lowering of unsafeAtomicAdd (expect global_atomic_add_f32,
// not a CAS loop). It is never launched.

#define N_NODES 200000
#define N_DIM   64
#define NFLOAT  (N_NODES * N_DIM)   // 12,800,000 floats per node-matrix
#define NVEC4   (NFLOAT / 4)        // 3,200,000 float4

#define SPMM_BLOCK 256
#define WPB (SPMM_BLOCK / 32)       // waves per block (wave32)
#define CHUNK 128                   // edges per staging chunk (512B per array)

// ---------------- atomic lowering probe (never launched) ----------------

__global__ void k_atomic_probe(float* p, float v) {
  unsafeAtomicAdd(p + threadIdx.x, v);
}

// ---------------- SpMM: y[rows[e]] += vals[e] * x[cols[e]] ----------------

__global__ __launch_bounds__(SPMM_BLOCK)
void k_spmm(const float* __restrict__ x,
            const float* __restrict__ vals,
            const int* __restrict__ rows,
            const int* __restrict__ cols,
            float* __restrict__ y,
            int n_edges) {
  // Per-wave staging: [wave][double-buffer][rows/cols/vals][128 edges]
  __shared__ int sm[WPB][2][3][CHUNK];

  const int lane    = threadIdx.x & 31;
  const int wid     = threadIdx.x >> 5;
  const int nwaves  = gridDim.x * WPB;
  const int gwave   = blockIdx.x * WPB + wid;
  const int nchunks = (n_edges + CHUNK - 1) >> 7;

  // LDS byte offsets (low 32 bits of flat shared address == LDS offset).
  uint32_t ldsBase[2][3];
#pragma unroll
  for (int b = 0; b < 2; ++b)
#pragma unroll
    for (int a = 0; a < 3; ++a)
      ldsBase[b][a] = (uint32_t)(uintptr_t)&sm[wid][b][a][0];

  // Stage one 128-edge chunk asynchronously. Full chunks: 3 x b128 async
  // instructions (each moves 512B across the wave). Returns true if exactly
  // 3 async ops were issued (enables the pipelined asynccnt<=3 wait).
  auto prefetch = [&](int chunk, int b) -> bool {
    const int ebase = chunk << 7;
    if (ebase + CHUNK <= n_edges) {
      const uint32_t seg  = (uint32_t)lane << 4;          // 16B per lane
      const uint32_t goff = ((uint32_t)ebase << 2) + seg; // byte offset
      asm volatile("global_load_async_to_lds_b128 %0, %1, %2"
                   :: "v"(ldsBase[b][0] + seg), "v"(goff), "s"(rows) : "memory");
      asm volatile("global_load_async_to_lds_b128 %0, %1, %2"
                   :: "v"(ldsBase[b][1] + seg), "v"(goff), "s"(cols) : "memory");
      asm volatile("global_load_async_to_lds_b128 %0, %1, %2"
                   :: "v"(ldsBase[b][2] + seg), "v"(goff), "s"(vals) : "memory");
      return true;
    }
    // tail chunk: guarded per-lane dword staging
#pragma unroll
    for (int s = 0; s < 4; ++s) {
      const int e = ebase + (s << 5) + lane;
      if (e < n_edges) {
        const uint32_t d    = (uint32_t)((s << 5) + lane) << 2;
        const uint32_t goff = (uint32_t)e << 2;
        asm volatile("global_load_async_to_lds_b32 %0, %1, %2"
                     :: "v"(ldsBase[b][0] + d), "v"(goff), "s"(rows) : "memory");
        asm volatile("global_load_async_to_lds_b32 %0, %1, %2"
                     :: "v"(ldsBase[b][1] + d), "v"(goff), "s"(cols) : "memory");
        asm volatile("global_load_async_to_lds_b32 %0, %1, %2"
                     :: "v"(ldsBase[b][2] + d), "v"(goff), "s"(vals) : "memory");
      }
    }
    return false;
  };

  int chunk = gwave;
  if (chunk < nchunks) (void)prefetch(chunk, 0);

  const int sub = lane >> 4;        // which of the 2 edges this iteration
  const int l16 = lane & 15;        // 4-float group within the 64-dim row
  const float* xg = x + (l16 << 2); // lane-fixed dim offset
  float*       yg = y + (l16 << 2);

  int it = 0;
  for (; chunk < nchunks; chunk += nwaves, ++it) {
    const int buf = it & 1;
    const int nxt = chunk + nwaves;
    bool pipelined = false;
    if (nxt < nchunks) pipelined = prefetch(nxt, buf ^ 1);
    if (pipelined) {
      // 3 outstanding ops belong to the next chunk; older (this chunk's 3)
      // have completed since async loads complete in order.
      asm volatile("s_wait_asynccnt 0x3" ::: "memory");
    } else {
      asm volatile("s_wait_asynccnt 0x0" ::: "memory");
    }

    const int base = chunk << 7;
    if (base + CHUNK <= n_edges) {
      // steady state: no per-edge guard
#pragma unroll 8
      for (int j = 0; j < CHUNK; j += 2) {
        const int le = j + sub;
        const int   r = sm[wid][buf][0][le];
        const int   c = sm[wid][buf][1][le];
        const float v = __int_as_float(sm[wid][buf][2][le]);
        const float4 xv = *(const float4*)(xg + ((size_t)(uint32_t)c << 6));
        float* yr = yg + ((size_t)(uint32_t)r << 6);
        unsafeAtomicAdd(yr + 0, v * xv.x);
        unsafeAtomicAdd(yr + 1, v * xv.y);
        unsafeAtomicAdd(yr + 2, v * xv.z);
        unsafeAtomicAdd(yr + 3, v * xv.w);
      }
    } else {
      const int rem = n_edges - base;
      for (int j = 0; j < CHUNK; j += 2) {
        const int le = j + sub;
        if (le < rem) {
          const int   r = sm[wid][buf][0][le];
          const int   c = sm[wid][buf][1][le];
          const float v = __int_as_float(sm[wid][buf][2][le]);
          const float4 xv = *(const float4*)(xg + ((size_t)(uint32_t)c << 6));
          float* yr = yg + ((size_t)(uint32_t)r << 6);
          unsafeAtomicAdd(yr + 0, v * xv.x);
          unsafeAtomicAdd(yr + 1, v * xv.y);
          unsafeAtomicAdd(yr + 2, v * xv.z);
          unsafeAtomicAdd(yr + 3, v * xv.w);
        }
      }
    }
  }
}

// ---------------- elementwise helpers ----------------

__global__ void k_init(const float4* __restrict__ emb, float4* __restrict__ out,
                       float4* __restrict__ z0, float4* __restrict__ z1) {
  int i = blockIdx.x * blockDim.x + threadIdx.x;
  if (i < NVEC4) {
    out[i] = emb[i];
    float4 z; z.x = 0.f; z.y = 0.f; z.z = 0.f; z.w = 0.f;
    z0[i] = z;
    z1[i] = z;
  }
}

__global__ void k_acc(float4* __restrict__ out, const float4* __restrict__ h,
                      float4* __restrict__ z) {
  int i = blockIdx.x * blockDim.x + threadIdx.x;
  if (i < NVEC4) {
    float4 o = out[i];
    float4 a = h[i];
    o.x += a.x; o.y += a.y; o.z += a.z; o.w += a.w;
    out[i] = o;
    if (z != nullptr) {
      float4 zz; zz.x = 0.f; zz.y = 0.f; zz.z = 0.f; zz.w = 0.f;
      z[i] = zz;
    }
  }
}

__global__ void k_final(float4* __restrict__ out, const float4* __restrict__ h) {
  int i = blockIdx.x * blockDim.x + threadIdx.x;
  if (i < NVEC4) {
    float4 o = out[i];
    float4 a = h[i];
    o.x = (o.x + a.x) * 0.25f;
    o.y = (o.y + a.y) * 0.25f;
    o.z = (o.z + a.z) * 0.25f;
    o.w = (o.w + a.w) * 0.25f;
    out[i] = o;
  }
}

// ---------------- driver ----------------

extern "C" void kernel_launch(void* const* d_in, const int* in_sizes, int n_in,
                              void* d_out, int out_size, void* d_ws, size_t ws_size,
                              hipStream_t stream) {
  const float* emb  = (const float*)d_in[0];
  const float* vals = (const float*)d_in[1];
  const int*   rows = (const int*)d_in[2];
  const int*   cols = (const int*)d_in[3];
  float* out = (float*)d_out;

  // Two ping-pong node matrices in workspace (2 * 51.2 MB).
  float* ws0 = (float*)d_ws;
  float* ws1 = ws0 + (size_t)NFLOAT;

  const int n_edges = in_sizes[1];

  dim3 eb(256), eg((NVEC4 + 255) / 256);
  dim3 sb(SPMM_BLOCK), sg(2048);   // 16384 waves -> deep atomic pipeline

  // out = emb; h buffers zeroed
  k_init<<<eg, eb, 0, stream>>>((const float4*)emb, (float4*)out,
                                (float4*)ws0, (float4*)ws1);
  // layer 1: h1 = A * emb
  k_spmm<<<sg, sb, 0, stream>>>(emb, vals, rows, cols, ws0, n_edges);
  k_acc<<<eg, eb, 0, stream>>>((float4*)out, (const float4*)ws0, (float4*)nullptr);
  // layer 2: h2 = A * h1
  k_spmm<<<sg, sb, 0, stream>>>(ws0, vals, rows, cols, ws1, n_edges);
  k_acc<<<eg, eb, 0, stream>>>((float4*)out, (const float4*)ws1, (float4*)ws0);
  // layer 3: h3 = A * h2
  k_spmm<<<sg, sb, 0, stream>>>(ws1, vals, rows, cols, ws0, n_edges);
  // out = (emb + h1 + h2 + h3) / 4
  k_final<<<eg, eb, 0, stream>>>((float4*)out, (const float4*)ws0);
}